// CARAFE_40389872452091
// MI455X (gfx1250) — compile-verified
//
#include <hip/hip_runtime.h>

typedef __attribute__((ext_vector_type(2))) float v2f;
typedef __attribute__((ext_vector_type(8))) float v8f;
typedef __attribute__((ext_vector_type(4))) int   v4i;

#define C_ 256
#define H_ 128
#define W_ 128
#define SH_ 256
#define SW_ 256
#define NK_ 25      // 5x5 taps

#define FCOLS 68    // staged feature cols: 128px block -> 64 src cols + 4 halo
#define FSTR  69    // padded LDS col stride (16-lane stride 345 dwords, odd -> conflict-free)

#if defined(__has_builtin)
#if __has_builtin(__builtin_amdgcn_global_load_async_to_lds_b128)
#define CARAFE_ASYNC_LDS 1
#endif
#endif

// Block (256 thr = 8 waves): 16 channels x 2 output rows (2ih, 2ih+1) x 128 px.
// Masks staged to LDS via GLOBAL_LOAD_ASYNC_TO_LDS_B128 (ASYNCcnt path);
// features staged synchronously (boundary clamp/zero transform + 4B-aligned
// padded stride). Each wave then runs two chained WMMA f32 16x16x4 series.
__global__ __launch_bounds__(256) void carafe_wmma_lds(
    const float* __restrict__ feat,   // (2,256,128,128)
    const float* __restrict__ mask,   // (2,25,256,256)
    float* __restrict__ out)          // (2,256,256,256)
{
    __shared__ float sfeat[16 * 5 * FSTR];   // [ch][di][col] 22.1 KB
    __shared__ float smask[2 * NK_ * 128];   // [row][tap][px] 25.6 KB

    const int t = threadIdx.x;
    const int b = blockIdx.x;

    const int owb_i = b & 1;            // 0/1 -> px block 0..127 / 128..255
    const int ihb   = (b >> 1) & 127;   // input row -> output rows 2ihb, 2ihb+1
    const int ct    = (b >> 8) & 15;    // channel tile
    const int n     = b >> 12;          // batch

    const int owb = owb_i << 7;         // 0 or 128
    const int iwb = owb >> 1;           // 0 or 64
    const int c0  = ct << 4;
    const int oh0 = ihb << 1;
    const int colbase = iwb - 2;        // global col of LDS col 0
    const int rowbase = ihb - 2;        // global row of LDS row 0

    // ---- stage masks: 1600 x 16B contiguous, aligned chunks
#ifdef CARAFE_ASYNC_LDS
    typedef v4i __attribute__((address_space(1))) gv4i;
    typedef v4i __attribute__((address_space(3))) sv4i;
    for (int idx4 = t; idx4 < 2 * NK_ * 32; idx4 += 256) {
        const int px4 = idx4 & 31;      // float4 within 128-px run
        const int tr  = idx4 >> 5;      // row*25 + tap
        const int tap = tr % NK_;
        const int row = tr / NK_;
        const float* g = mask + ((size_t)(n * NK_ + tap)) * (SH_ * SW_)
                              + (size_t)(oh0 + row) * SW_ + owb + (px4 << 2);
        __builtin_amdgcn_global_load_async_to_lds_b128(
            (gv4i*)g, (sv4i*)(smask + (idx4 << 2)), 0, 0);
    }
#else
    for (int idx = t; idx < 2 * NK_ * 128; idx += 256) {
        const int px  = idx & 127;
        const int tr  = idx >> 7;       // row*25 + tap
        const int tap = tr % NK_;
        const int row = tr / NK_;
        smask[idx] = mask[((size_t)(n * NK_ + tap)) * (SH_ * SW_)
                          + (size_t)(oh0 + row) * SW_ + owb + px];
    }
#endif

    // ---- stage features: coalesced 68-float runs, clamp + zero-select (no branches)
    for (int idx = t; idx < 16 * 5 * FCOLS; idx += 256) {
        const int col = idx % FCOLS;
        const int rc  = idx / FCOLS;    // 0..79
        const int row = rc % 5;
        const int ch  = rc / 5;
        const int gcol = colbase + col;
        const int grow = rowbase + row;
        const bool ok = ((unsigned)gcol < (unsigned)W_) &&
                        ((unsigned)grow < (unsigned)H_);
        const int gci = min(max(gcol, 0), W_ - 1);
        const int gri = min(max(grow, 0), H_ - 1);
        const float v =
            feat[((size_t)(n * C_ + c0 + ch)) * (H_ * W_) + gri * W_ + gci];
        sfeat[ch * (5 * FSTR) + row * FSTR + col] = ok ? v : 0.f;
    }

#ifdef CARAFE_ASYNC_LDS
    // drain this wave's async copies, then rendezvous with the other waves
#if __has_builtin(__builtin_amdgcn_s_wait_asynccnt)
    __builtin_amdgcn_s_wait_asynccnt(0);
#else
    asm volatile("s_wait_asynccnt 0x0" ::: "memory");
#endif
#endif
    __syncthreads();

    // ---- per-wave WMMA: 16ch x 16px tile, both output rows
    const int lane = t & 31;
    const int w    = t >> 5;            // wave = px tile 0..7
    const int half = lane >> 4;         // A: K base +2 ; D: M base +8
    const int l16  = lane & 15;         // A: channel row ; B: pixel col
    const int phalf = l16 >> 1;         // pixel -> src col offset
    const int cA = w << 3;              // wave's LDS col for K=0

    const float* fA  = sfeat + l16 * (5 * FSTR);
    const float* mB0 = smask + (w << 4) + l16;             // row 0
    const float* mB1 = smask + NK_ * 128 + (w << 4) + l16; // row 1

    v8f acc0 = {0.f, 0.f, 0.f, 0.f, 0.f, 0.f, 0.f, 0.f};
    v8f acc1 = {0.f, 0.f, 0.f, 0.f, 0.f, 0.f, 0.f, 0.f};

    #pragma unroll
    for (int di = 0; di < 5; ++di) {
        #pragma unroll
        for (int j = 0; j < 3; ++j) {
            const int k0 = 4 * j + 2 * half;   // ISA 16x4 f32 A/B lane K-slots
            const int k1 = k0 + 1;

            v2f a;
            a.x = fA[di * FSTR + cA + k0];
            a.y = fA[di * FSTR + cA + k1];

            // banded B: B[k,p] = mask[di*5+dj], dj = k - p/2, zero outside band
            const int dj0 = k0 - phalf;
            const int dj1 = k1 - phalf;
            const bool v0 = (unsigned)dj0 < 5u;
            const bool v1 = (unsigned)dj1 < 5u;
            const int t0 = di * 5 + min(max(dj0, 0), 4);
            const int t1 = di * 5 + min(max(dj1, 0), 4);

            v2f b0, b1;
            const float x0 = mB0[t0 << 7], x1 = mB0[t1 << 7];
            const float y0 = mB1[t0 << 7], y1 = mB1[t1 << 7];
            b0.x = v0 ? x0 : 0.f;  b0.y = v1 ? x1 : 0.f;
            b1.x = v0 ? y0 : 0.f;  b1.y = v1 ? y1 : 0.f;

            acc0 = __builtin_amdgcn_wmma_f32_16x16x4_f32(
                false, a, false, b0, (short)0, acc0, false, false);
            acc1 = __builtin_amdgcn_wmma_f32_16x16x4_f32(
                false, a, false, b1, (short)0, acc1, false, false);
        }
    }

    // ---- store: D VGPR r -> channel c0 + r + 8*half, px = l16; coalesced in px
    float* ob = out + (((size_t)(n * C_ + c0 + 8 * half)) * SH_ + oh0) * SW_
                    + owb + (w << 4) + l16;
    #pragma unroll
    for (int r = 0; r < 8; ++r) {
        ob[(size_t)r * (SH_ * SW_)]       = acc0[r];
        ob[(size_t)r * (SH_ * SW_) + SW_] = acc1[r];
    }
}

extern "C" void kernel_launch(void* const* d_in, const int* in_sizes, int n_in,
                              void* d_out, int out_size, void* d_ws, size_t ws_size,
                              hipStream_t stream) {
    const float* feat = (const float*)d_in[0];   // 2*256*128*128
    const float* mask = (const float*)d_in[1];   // 2*25*256*256
    float* out = (float*)d_out;                  // 2*256*256*256

    // 2 batch * 16 ctiles * 128 input rows * 2 px-blocks = 8192 blocks
    carafe_wmma_lds<<<8192, 256, 0, stream>>>(feat, mask, out);
}